// DualPathMambaBlocks_84413287235929
// MI455X (gfx1250) — compile-verified
//
#include <hip/hip_runtime.h>
#include <hip/hip_bf16.h>
#include <math.h>

// ---- problem constants (match reference) ----
#define DM     256      // D_MODEL
#define DI     512      // D_INNER
#define DSn    16       // D_STATE
#define DTR    16       // DT_RANK
#define KCONV  4        // D_CONV
#define NDEPTH 4
#define BB     4
#define TT     8
#define HD_    16
#define WD_    16
#define HS_    8
#define WS_    8
#define LDTOK  (TT*HD_*WD_)   // 2048 detail tokens per batch
#define LSTOK  (TT*HS_*WS_)   // 512 semantic tokens per batch
#define NDROWS (BB*LDTOK)     // 8192
#define NSROWS (BB*LSTOK)     // 2048
#define EPSF   1e-5f

typedef float v2f __attribute__((ext_vector_type(2)));
typedef float v8f __attribute__((ext_vector_type(8)));

// ---- fragment helpers: one 8-wide k-chunk = two WMMA k-steps ----
template<int NSUB>
__device__ __forceinline__ void load_chunk(const float* arow, const float* const* wrow,
                                           int koff, v2f& a0, v2f& a1, v2f* b0, v2f* b1)
{
    a0 = *(const v2f*)(arow + koff);
    a1 = *(const v2f*)(arow + koff + 4);
    #pragma unroll
    for (int j = 0; j < NSUB; ++j) {
        b0[j] = *(const v2f*)(wrow[j] + koff);
        b1[j] = *(const v2f*)(wrow[j] + koff + 4);
    }
}

template<int NSUB>
__device__ __forceinline__ void do_wmma(const v2f& a0, const v2f& a1,
                                        const v2f* b0, const v2f* b1, v8f* acc)
{
    #pragma unroll
    for (int j = 0; j < NSUB; ++j)
        acc[j] = __builtin_amdgcn_wmma_f32_16x16x4_f32(
            false, a0, false, b0[j], (short)0, acc[j], false, false);
    #pragma unroll
    for (int j = 0; j < NSUB; ++j)
        acc[j] = __builtin_amdgcn_wmma_f32_16x16x4_f32(
            false, a1, false, b1[j], (short)0, acc[j], false, false);
}

// =====================================================================
// WMMA f32 GEMM:  C[M,N] = act(A[M,K] @ W[N,K]^T + bias)
// One wave computes a 16 x (16*NSUB) tile of C; NSUB is compile-time so
// the j-loops are branch-free (EXEC stays all-ones around v_wmma).
// Two-phase software pipeline over 8-wide k-chunks (requires K % 16 == 0):
// chunk c+1 is loaded into the *other* phase's registers before the WMMAs
// of chunk c issue, so loads overlap math under partial s_wait_loadcnt.
// ACT: 0=none, 1=+bias, 2=+bias then exact GELU, 3=+bias then softplus
// =====================================================================
template<int ACT, int NSUB>
__global__ __launch_bounds__(256)
void wmma_gemm_f32(const float* __restrict__ A, const float* __restrict__ W,
                   const float* __restrict__ bias, float* __restrict__ C,
                   int M, int N, int K, int lda, int ldw, int ldc)
{
    const int wave    = blockIdx.x * (blockDim.x >> 5) + (threadIdx.x >> 5);
    const int lane    = threadIdx.x & 31;
    const int tiles_n = N / (16 * NSUB);
    const int tiles_m = M >> 4;
    if (wave >= tiles_m * tiles_n) return;           // wave-uniform exit

    const int m0   = (wave / tiles_n) << 4;
    const int n0   = (wave % tiles_n) * (16 * NSUB);
    const int half = lane >> 4;                      // which K-pair this half-wave owns
    const int lm   = lane & 15;

    v8f zero;
    #pragma unroll
    for (int e = 0; e < 8; ++e) zero[e] = 0.f;
    v8f acc[NSUB];
    #pragma unroll
    for (int j = 0; j < NSUB; ++j) acc[j] = zero;

    const float* arow = A + (size_t)(m0 + lm) * lda + 2 * half;
    const float* wrow[NSUB];
    #pragma unroll
    for (int j = 0; j < NSUB; ++j)
        wrow[j] = W + (size_t)(n0 + j * 16 + lm) * ldw + 2 * half;

    // even / odd phase fragment registers (disjoint -> no copies at swap)
    v2f aE0, aE1, bE0[NSUB], bE1[NSUB];
    v2f aO0, aO1, bO0[NSUB], bO1[NSUB];

    load_chunk<NSUB>(arow, wrow, 0, aE0, aE1, bE0, bE1);
    int k0 = 0;
    for (; k0 + 16 < K; k0 += 16) {
        load_chunk<NSUB>(arow, wrow, k0 + 8,  aO0, aO1, bO0, bO1);
        do_wmma<NSUB>(aE0, aE1, bE0, bE1, acc);
        load_chunk<NSUB>(arow, wrow, k0 + 16, aE0, aE1, bE0, bE1);
        do_wmma<NSUB>(aO0, aO1, bO0, bO1, acc);
    }
    load_chunk<NSUB>(arow, wrow, k0 + 8, aO0, aO1, bO0, bO1);
    do_wmma<NSUB>(aE0, aE1, bE0, bE1, acc);
    do_wmma<NSUB>(aO0, aO1, bO0, bO1, acc);

    #pragma unroll
    for (int j = 0; j < NSUB; ++j) {
        const int n = n0 + j * 16 + lm;
        const float bv = (ACT >= 1) ? bias[n] : 0.f;
        #pragma unroll
        for (int r = 0; r < 8; ++r) {
            const int m = m0 + r + half * 8;
            float v = acc[j][r];
            if (ACT >= 1) v += bv;
            if (ACT == 2) v = 0.5f * v * (1.f + erff(v * 0.70710678118654752f)); // exact GELU
            if (ACT == 3) v = (v > 20.f) ? v : log1pf(__expf(v));                // softplus
            C[(size_t)m * ldc + n] = v;
        }
    }
}

template<int ACT, int NSUB>
static inline void launch_gemm(const float* A, const float* W, const float* bias, float* C,
                               int M, int N, int K, int lda, int ldw, int ldc,
                               hipStream_t stream)
{
    int tiles  = (M / 16) * (N / (16 * NSUB));
    int blocks = (tiles + 7) / 8;                    // 8 waves per 256-thread block
    wmma_gemm_f32<ACT, NSUB><<<blocks, 256, 0, stream>>>(A, W, bias, C, M, N, K, lda, ldw, ldc);
}

// ---- (B,T,C,H,W) -> (B, T*H*W, C) token flatten ----
__global__ void flatten_kernel(const float* __restrict__ x, float* __restrict__ out, int H, int W)
{
    int i = blockIdx.x * blockDim.x + threadIdx.x;   // B*L*DM elements
    int c = i & (DM - 1); int row = i >> 8;
    int HW = H * W; int L = TT * HW;
    int b = row / L; int lr = row - b * L;
    int t = lr / HW; int hw = lr - t * HW;
    int h = hw / W;  int w = hw - h * W;
    out[i] = x[((((size_t)b * TT + t) * DM + c) * H + h) * W + w];
}

// ---- res = h (+ res);  hn = rmsnorm(res) * w ----
__global__ __launch_bounds__(256)
void add_rmsnorm_kernel(const float* __restrict__ h, float* res,
                        const float* __restrict__ w, float* __restrict__ hn, int hasres)
{
    __shared__ float red[256];
    int row = blockIdx.x, c = threadIdx.x;
    size_t idx = (size_t)row * DM + c;
    float v = h[idx] + (hasres ? res[idx] : 0.f);
    res[idx] = v;
    red[c] = v * v; __syncthreads();
    for (int s = 128; s > 0; s >>= 1) { if (c < s) red[c] += red[c + s]; __syncthreads(); }
    hn[idx] = v * rsqrtf(red[0] * (1.f / DM) + EPSF) * w[c];
}

// ---- causal depthwise conv (k=4, per-batch) + SiLU; x = xz[:, 0:DI] ----
__global__ void conv_silu_kernel(const float* __restrict__ xz, const float* __restrict__ cw,
                                 const float* __restrict__ cb, float* __restrict__ out, int L)
{
    int i = blockIdx.x * blockDim.x + threadIdx.x;   // B*L*DI
    int d = i & (DI - 1); int row = i >> 9;
    int l = row % L;
    float acc = cb[d];
    #pragma unroll
    for (int j = 0; j < KCONV; ++j) {
        int ll = l - (KCONV - 1) + j;
        if (ll >= 0)
            acc += cw[d * KCONV + j] * xz[(size_t)(row - (KCONV - 1 - j)) * (2 * DI) + d];
    }
    out[(size_t)row * DI + d] = acc / (1.f + __expf(-acc));   // silu
}

// ---- selective scan fused with +x*D and *silu(z); one thread per (b,d) ----
// NOTE: g may alias dt (each thread reads dt[row,d] before writing g[row,d]).
__global__ void scan_gate_kernel(const float* __restrict__ xc, const float* dt,
                                 const float* __restrict__ xdbl, const float* __restrict__ xz,
                                 const float* __restrict__ Alog, const float* __restrict__ Dp,
                                 float* g, int L)
{
    int i = blockIdx.x * blockDim.x + threadIdx.x;   // B*DI threads
    int d = i & (DI - 1); int b = i >> 9;
    float A[DSn], h[DSn];
    #pragma unroll
    for (int s = 0; s < DSn; ++s) { A[s] = -__expf(Alog[d * DSn + s]); h[s] = 0.f; }
    const float Dd = Dp[d];
    for (int l = 0; l < L; ++l) {
        size_t row = (size_t)b * L + l;
        float xt  = xc[row * DI + d];
        float dtt = dt[row * DI + d];
        const float* br = xdbl + row * 48 + DTR;          // B(t)
        const float* cr = xdbl + row * 48 + DTR + DSn;    // C(t)
        float dx = dtt * xt;
        float y = 0.f;
        #pragma unroll
        for (int s = 0; s < DSn; ++s) {
            h[s] = h[s] * __expf(dtt * A[s]) + dx * br[s];
            y += h[s] * cr[s];
        }
        float z = xz[row * (2 * DI) + DI + d];
        g[row * DI + d] = (y + xt * Dd) * (z / (1.f + __expf(-z)));
    }
}

// ---- build [hd | hs_upsampled] feature concat at detail resolution ----
__global__ void cat_kernel(const float* __restrict__ hd, const float* __restrict__ hs,
                           float* __restrict__ cat)
{
    int i = blockIdx.x * blockDim.x + threadIdx.x;   // NDROWS * 2*DM
    int c = i & (2 * DM - 1); int row = i >> 9;
    if (c < DM) {
        cat[i] = hd[(size_t)row * DM + c];
    } else {
        int b = row / LDTOK; int lr = row - b * LDTOK;
        int t = lr >> 8; int hw = lr & 255;
        int hh = (hw >> 4) >> 1; int ww = (hw & 15) >> 1;
        int ls = t * (HS_ * WS_) + hh * WS_ + ww;
        cat[i] = hs[((size_t)b * LSTOK + ls) * DM + (c - DM)];
    }
}

// ---- layernorm in place (row = 256) ----
__global__ __launch_bounds__(256)
void layernorm_kernel(float* x, const float* __restrict__ w, const float* __restrict__ b)
{
    __shared__ float r1[256], r2[256];
    int row = blockIdx.x, c = threadIdx.x;
    size_t idx = (size_t)row * DM + c;
    float v = x[idx];
    r1[c] = v; r2[c] = v * v; __syncthreads();
    for (int s = 128; s > 0; s >>= 1) {
        if (c < s) { r1[c] += r1[c + s]; r2[c] += r2[c + s]; }
        __syncthreads();
    }
    float m = r1[0] * (1.f / DM);
    float var = r2[0] * (1.f / DM) - m * m;
    x[idx] = (v - m) * rsqrtf(var + EPSF) * w[c] + b[c];
}

__global__ void axpy_kernel(float* h, const float* __restrict__ fn,
                            const float* __restrict__ alpha, int ai)
{
    int i = blockIdx.x * blockDim.x + threadIdx.x;
    h[i] += alpha[ai] * fn[i];
}

// ---- hs += beta * avg(2x2 block of fn) ----
__global__ void hs_update_kernel(float* hs, const float* __restrict__ fn,
                                 const float* __restrict__ beta, int bi)
{
    int i = blockIdx.x * blockDim.x + threadIdx.x;   // NSROWS*DM
    int c = i & (DM - 1); int row = i >> 8;
    int b = row / LSTOK; int lr = row - b * LSTOK;
    int t = lr >> 6; int q = lr & 63;
    int hh = q >> 3; int ww = q & 7;
    int base = b * LDTOK + t * (HD_ * WD_);
    float s = 0.f;
    #pragma unroll
    for (int dy = 0; dy < 2; ++dy)
        #pragma unroll
        for (int dx = 0; dx < 2; ++dx) {
            int ld = base + (2 * hh + dy) * WD_ + (2 * ww + dx);
            s += fn[(size_t)ld * DM + c];
        }
    hs[i] += beta[bi] * 0.25f * s;
}

// ---- final rmsnorm(h+res) and un-flatten back to (B,T,C,H,W) ----
__global__ __launch_bounds__(256)
void final_rmsnorm_unflat_kernel(const float* __restrict__ h, const float* __restrict__ res,
                                 const float* __restrict__ w, float* __restrict__ out,
                                 int H, int W)
{
    __shared__ float red[256];
    int row = blockIdx.x, c = threadIdx.x;
    size_t idx = (size_t)row * DM + c;
    float v = h[idx] + res[idx];
    red[c] = v * v; __syncthreads();
    for (int s = 128; s > 0; s >>= 1) { if (c < s) red[c] += red[c + s]; __syncthreads(); }
    float nrm = rsqrtf(red[0] * (1.f / DM) + EPSF);
    int HW = H * W; int L = TT * HW;
    int b = row / L; int lr = row - b * L;
    int t = lr / HW; int hw = lr - t * HW;
    int hh = hw / W; int ww = hw - hh * W;
    out[((((size_t)b * TT + t) * DM + c) * H + hh) * W + ww] = v * nrm * w[c];
}

__global__ void write_scalar_kernel(float* p, float v)
{
    if (blockIdx.x == 0 && threadIdx.x == 0) p[0] = v;
}

// =====================================================================
extern "C" void kernel_launch(void* const* d_in, const int* in_sizes, int n_in,
                              void* d_out, int out_size, void* d_ws, size_t ws_size,
                              hipStream_t stream)
{
    (void)in_sizes; (void)n_in; (void)out_size; (void)ws_size;

    // params tree flattened by sorted dict keys:
    const float* x_detail   = (const float*)d_in[0];
    const float* x_semantic = (const float*)d_in[1];
    const float* det_D      = (const float*)d_in[2];
    const float* det_cb     = (const float*)d_in[3];
    const float* det_cw     = (const float*)d_in[4];
    const float* det_dtb    = (const float*)d_in[5];
    const float* det_dtw    = (const float*)d_in[6];
    const float* det_nw     = (const float*)d_in[7];
    const float* det_ow     = (const float*)d_in[8];
    const float* det_xw     = (const float*)d_in[9];
    const float* it_alpha   = (const float*)d_in[10];
    const float* it_b1      = (const float*)d_in[11];
    const float* it_b2      = (const float*)d_in[12];
    const float* it_beta    = (const float*)d_in[13];
    const float* it_lnb     = (const float*)d_in[14];
    const float* it_lnw     = (const float*)d_in[15];
    const float* it_w1      = (const float*)d_in[16];
    const float* it_w2      = (const float*)d_in[17];
    const float* normf_d    = (const float*)d_in[18];
    const float* normf_s    = (const float*)d_in[19];
    const float* sem_Alog   = (const float*)d_in[20];
    const float* sem_D      = (const float*)d_in[21];
    const float* sem_cb     = (const float*)d_in[22];
    const float* sem_cw     = (const float*)d_in[23];
    const float* sem_dtb    = (const float*)d_in[24];
    const float* sem_dtw    = (const float*)d_in[25];
    const float* sem_inw    = (const float*)d_in[26];   // shared in_proj (both paths)
    const float* sem_nw     = (const float*)d_in[27];
    const float* sem_ow     = (const float*)d_in[28];
    const float* sem_xw     = (const float*)d_in[29];

    // workspace layout (floats)
    float* ws    = (float*)d_ws;
    float* hd    = ws; ws += (size_t)NDROWS * DM;
    float* rd    = ws; ws += (size_t)NDROWS * DM;
    float* hsb   = ws; ws += (size_t)NSROWS * DM;
    float* rsb   = ws; ws += (size_t)NSROWS * DM;
    float* hn    = ws; ws += (size_t)NDROWS * DM;
    float* xz    = ws; ws += (size_t)NDROWS * 2 * DI;   // reused for fusion temporaries
    float* xconv = ws; ws += (size_t)NDROWS * DI;
    float* xdbl  = ws; ws += (size_t)NDROWS * 48;
    float* dtg   = ws; ws += (size_t)NDROWS * DI;       // dt, then gated scan output
    float* catb  = xz;                                   // NDROWS*512
    float* t1    = xz + (size_t)NDROWS * 2 * DM;         // NDROWS*256
    float* fnb   = t1 + (size_t)NDROWS * DM;             // NDROWS*256

    flatten_kernel<<<(NDROWS * DM) / 256, 256, 0, stream>>>(x_detail, hd, HD_, WD_);
    flatten_kernel<<<(NSROWS * DM) / 256, 256, 0, stream>>>(x_semantic, hsb, HS_, WS_);

    auto run_path = [&](float* h, float* res, int L, int layer,
                        const float* nw, const float* cw, const float* cb,
                        const float* xw, const float* dtw, const float* dtb,
                        const float* Dp, const float* ow) {
        const int Nrow = BB * L;
        add_rmsnorm_kernel<<<Nrow, 256, 0, stream>>>(h, res, nw + layer * DM, hn, layer > 0);
        // xz = hn @ in_w.T                (Nrow,256) x (1024,256)^T
        launch_gemm<0, 4>(hn, sem_inw + (size_t)layer * 2 * DI * DM, nullptr, xz,
                          Nrow, 2 * DI, DM, DM, DM, 2 * DI, stream);
        conv_silu_kernel<<<(Nrow * DI) / 256, 256, 0, stream>>>(
            xz, cw + (size_t)layer * DI * KCONV, cb + (size_t)layer * DI, xconv, L);
        // xdbl = x @ x_proj.T             (Nrow,512) x (48,512)^T  -> one 48-wide tile (NSUB=3)
        launch_gemm<0, 3>(xconv, xw + (size_t)layer * 48 * DI, nullptr, xdbl,
                          Nrow, 48, DI, DI, DI, 48, stream);
        // dt = softplus(xdbl[:, :16] @ dtw.T + dtb)
        launch_gemm<3, 4>(xdbl, dtw + (size_t)layer * DI * DTR, dtb + (size_t)layer * DI, dtg,
                          Nrow, DI, DTR, 48, DTR, DI, stream);
        scan_gate_kernel<<<(BB * DI) / 256, 256, 0, stream>>>(
            xconv, dtg, xdbl, xz, sem_Alog + (size_t)layer * DI * DSn,
            Dp + (size_t)layer * DI, dtg, L);
        // h = g @ out_w.T                 (Nrow,512) x (256,512)^T
        launch_gemm<0, 4>(dtg, ow + (size_t)layer * DM * DI, nullptr, h,
                          Nrow, DM, DI, DI, DI, DM, stream);
    };

    for (int i = 0; i < NDEPTH; ++i) {
        run_path(hd, rd, LDTOK, i, det_nw, det_cw, det_cb, det_xw, det_dtw, det_dtb, det_D, det_ow);
        run_path(hsb, rsb, LSTOK, i, sem_nw, sem_cw, sem_cb, sem_xw, sem_dtw, sem_dtb, sem_D, sem_ow);
        if (i < NDEPTH - 1) {
            cat_kernel<<<(NDROWS * 2 * DM) / 256, 256, 0, stream>>>(hd, hsb, catb);
            launch_gemm<2, 4>(catb, it_w1 + (size_t)i * DM * 2 * DM, it_b1 + (size_t)i * DM, t1,
                              NDROWS, DM, 2 * DM, 2 * DM, 2 * DM, DM, stream);
            launch_gemm<1, 4>(t1, it_w2 + (size_t)i * DM * DM, it_b2 + (size_t)i * DM, fnb,
                              NDROWS, DM, DM, DM, DM, DM, stream);
            layernorm_kernel<<<NDROWS, 256, 0, stream>>>(fnb, it_lnw + (size_t)i * DM,
                                                         it_lnb + (size_t)i * DM);
            axpy_kernel<<<(NDROWS * DM) / 256, 256, 0, stream>>>(hd, fnb, it_alpha, i);
            hs_update_kernel<<<(NSROWS * DM) / 256, 256, 0, stream>>>(hsb, fnb, it_beta, i);
        }
    }

    float* out = (float*)d_out;
    final_rmsnorm_unflat_kernel<<<NDROWS, 256, 0, stream>>>(hd, rd, normf_d, out, HD_, WD_);
    final_rmsnorm_unflat_kernel<<<NSROWS, 256, 0, stream>>>(hsb, rsb, normf_s,
                                                            out + (size_t)NDROWS * DM, HS_, WS_);
    write_scalar_kernel<<<1, 1, 0, stream>>>(out + (size_t)NDROWS * DM + (size_t)NSROWS * DM, 0.0f);
}